// IMV_72146860638381
// MI455X (gfx1250) — compile-verified
//
#include <hip/hip_runtime.h>
#include <math.h>

// Problem constants (match reference setup_inputs)
constexpr int B  = 16;
constexpr int T2 = 2048;
constexpr int T1 = 512;
constexpr int F  = 128;

constexpr float DELTA = 10.0f;
constexpr float EPS = 1e-8f;
constexpr float THRESHOLD = 0.9f;
constexpr float STEEPNESS = 10.0f;

typedef __attribute__((ext_vector_type(2))) float v2f;
typedef __attribute__((ext_vector_type(8))) float v8f;

// ---------------------------------------------------------------------------
// Kernel 1: per (b,t) row — imv_proposal = dot(alpha_row, arange(T1)),
//           activity = sigmoid(10*(0.9 - cos(mels[t-1],mels[t]))) * am
// One wave (32 lanes) per row; lane-strided coalesced loads.
// ---------------------------------------------------------------------------
__global__ __launch_bounds__(256)
void act_prop_kernel(const float* __restrict__ mels,
                     const float* __restrict__ alpha,
                     const int*   __restrict__ mel_mask,
                     float* __restrict__ act_out,
                     float* __restrict__ ivp_out) {
  int wave = (blockIdx.x * blockDim.x + threadIdx.x) >> 5;
  int lane = threadIdx.x & 31;
  if (wave >= B * T2) return;
  int t = wave % T2;

  // imv_proposal: sum_s alpha[b,t,s] * s
  const float* arow = alpha + (size_t)wave * T1;
  float acc = 0.f;
#pragma unroll
  for (int i = 0; i < T1 / 32; ++i) {
    int s = lane + 32 * i;
    acc += arow[s] * (float)s;
  }
#pragma unroll
  for (int off = 16; off > 0; off >>= 1) acc += __shfl_down(acc, off, 32);

  // activity via cosine similarity of consecutive frames
  float a_act = 1.f;
  if (t != 0) {
    const float* x = mels + (size_t)(wave - 1) * F;
    const float* y = mels + (size_t)wave * F;
    float dot = 0.f, n1 = 0.f, n2 = 0.f;
#pragma unroll
    for (int i = 0; i < F / 32; ++i) {
      int f = lane + 32 * i;
      float xv = x[f], yv = y[f];
      dot += xv * yv; n1 += xv * xv; n2 += yv * yv;
    }
#pragma unroll
    for (int off = 16; off > 0; off >>= 1) {
      dot += __shfl_down(dot, off, 32);
      n1  += __shfl_down(n1,  off, 32);
      n2  += __shfl_down(n2,  off, 32);
    }
    if (lane == 0) {
      float cs = dot / (fmaxf(sqrtf(n1), 1e-12f) * fmaxf(sqrtf(n2), 1e-12f));
      a_act = 1.f / (1.f + expf(-STEEPNESS * (THRESHOLD - cs)));
    }
  }
  if (lane == 0) {
    float am = (mel_mask[wave] != 0) ? 1.f : 0.f;
    act_out[wave] = a_act * am;
    ivp_out[wave] = acc;
  }
}

// ---------------------------------------------------------------------------
// Kernel 2: per-batch gated cumsum (shfl inclusive scan) + rescale.
// One wave per batch. Writes final imv to ws and to d_out imv section.
// ---------------------------------------------------------------------------
__global__ __launch_bounds__(32)
void scan_kernel(const float* __restrict__ act,
                 const float* __restrict__ ivp,
                 const int*   __restrict__ mel_mask,
                 const int*   __restrict__ text_mask,
                 float* __restrict__ imv,
                 float* __restrict__ out_imv) {
  int b = blockIdx.x;
  int lane = threadIdx.x;

  // masked lengths
  int msum = 0;
#pragma unroll
  for (int i = 0; i < T2 / 32; ++i) msum += (mel_mask[b * T2 + lane + 32 * i] != 0);
  int tsum = 0;
#pragma unroll
  for (int i = 0; i < T1 / 32; ++i) tsum += (text_mask[b * T1 + lane + 32 * i] != 0);
#pragma unroll
  for (int off = 16; off > 0; off >>= 1) {
    msum += __shfl_down(msum, off, 32);
    tsum += __shfl_down(tsum, off, 32);
  }
  msum = __shfl(msum, 0, 32);
  tsum = __shfl(tsum, 0, 32);

  float carry = 0.f;
  for (int c = 0; c < T2 / 32; ++c) {
    int t = c * 32 + lane;
    int row = b * T2 + t;
    float am = (mel_mask[row] != 0) ? 1.f : 0.f;
    float d = (t == 0) ? 0.f : (ivp[row] - ivp[row - 1]);
    float eff = fminf(fmaxf(d * act[row], 0.f), 1.f);
    float v = eff;
#pragma unroll
    for (int off = 1; off < 32; off <<= 1) {
      float u = __shfl_up(v, off, 32);
      if (lane >= off) v += u;
    }
    imv[row] = (carry + v) * am;
    carry += __shfl(v, 31, 32);
  }
  __threadfence();
  __syncthreads();

  int last_idx = max(msum - 1, 0);
  float last_val = fmaxf(imv[b * T2 + last_idx], 1e-6f);
  float scale = fmaxf((float)tsum - 1.f, 0.f) / last_val;
  for (int c = 0; c < T2 / 32; ++c) {
    int t = c * 32 + lane;
    int row = b * T2 + t;
    float am = (mel_mask[row] != 0) ? 1.f : 0.f;
    float v = imv[row] * scale * am;
    imv[row] = v;
    out_imv[row] = v;
  }
}

// ---------------------------------------------------------------------------
// Kernel 3: per (b,t) softmax stats over T1: rowmax m and Z = sum exp(l - m).
// One wave per row.
// ---------------------------------------------------------------------------
__global__ __launch_bounds__(256)
void stats_kernel(const float* __restrict__ imv,
                  const int*   __restrict__ text_mask,
                  float* __restrict__ mrow,
                  float* __restrict__ zrow) {
  int wave = (blockIdx.x * blockDim.x + threadIdx.x) >> 5;
  int lane = threadIdx.x & 31;
  if (wave >= B * T2) return;
  int b = wave / T2;
  float iv = imv[wave];

  float loc[T1 / 32];
  float mx = -3.0e38f;
#pragma unroll
  for (int i = 0; i < T1 / 32; ++i) {
    int s = lane + 32 * i;
    float d = iv - (float)s;
    float l = (text_mask[b * T1 + s] != 0) ? (-DELTA * d * d) : -1e9f;
    loc[i] = l;
    mx = fmaxf(mx, l);
  }
#pragma unroll
  for (int off = 16; off > 0; off >>= 1) mx = fmaxf(mx, __shfl_down(mx, off, 32));
  mx = __shfl(mx, 0, 32);

  float z = 0.f;
#pragma unroll
  for (int i = 0; i < T1 / 32; ++i) z += expf(loc[i] - mx);
#pragma unroll
  for (int off = 16; off > 0; off >>= 1) z += __shfl_down(z, off, 32);
  if (lane == 0) { mrow[wave] = mx; zrow[wave] = z; }
}

// ---------------------------------------------------------------------------
// Kernel 4: block = (b, s-tile of 16). 8 waves, wave w owns f-tile w (16 cols).
// gamma tile [32t x 16s] computed once into LDS, consumed as WMMA A-operand
// (gamma^T, 16x4 f32) against mels B-operand (4x16 f32):
//   C[16s x 16f] += A x B  via v_wmma_f32_16x16x4_f32, 8 K-steps per chunk.
// Durations accumulate per-thread, reduced with LDS ds_add_f32.
// ---------------------------------------------------------------------------
__global__ __launch_bounds__(256)
void align_gemm_kernel(const float* __restrict__ mels,
                       const int*   __restrict__ mel_mask,
                       const int*   __restrict__ text_mask,
                       const float* __restrict__ imv,
                       const float* __restrict__ mrow,
                       const float* __restrict__ zrow,
                       float* __restrict__ out_aligned,
                       float* __restrict__ out_dur) {
  int b = blockIdx.x >> 5;        // 32 s-tiles per batch
  int stile = blockIdx.x & 31;
  int s0 = stile * 16;

  int tid  = threadIdx.x;
  int lane = tid & 31;
  int wavei = tid >> 5;           // f-tile index, 0..7

  __shared__ float gamma_lds[32][16];
  __shared__ float dur_lds[16];
  if (tid < 16) dur_lds[tid] = 0.f;

  int s_local = tid & 15;
  int tl0 = tid >> 4;             // 0..15
  int s_glob = s0 + s_local;
  float tmf = (text_mask[b * T1 + s_glob] != 0) ? 1.f : 0.f;

  int ftile = wavei;
  int n  = lane & 15;             // B/C column within tile
  int m  = lane & 15;             // A row (phoneme) within tile
  int kb = (lane >> 4) * 2;       // K pair selected by lane half (ISA A layout)

  float dur_acc = 0.f;
  v8f cacc = {};                  // C/D accumulator, 8 VGPRs

  for (int t0 = 0; t0 < T2; t0 += 32) {
    // --- compute gamma tile [32 t][16 s] into LDS (2 elements per thread) ---
#pragma unroll
    for (int half = 0; half < 2; ++half) {
      int tl = tl0 + half * 16;
      int row = b * T2 + t0 + tl;
      float amf = (mel_mask[row] != 0) ? 1.f : 0.f;
      float g = 0.f;
      if (tmf > 0.f && amf > 0.f) {
        float d = imv[row] - (float)s_glob;
        g = expf(-DELTA * d * d - mrow[row]) / zrow[row];
      }
      dur_acc += g;
      gamma_lds[tl][s_local] = g;
    }
    __syncthreads();

    // --- 8 K-steps of V_WMMA_F32_16X16X4_F32 per 32-t chunk ---
#pragma unroll
    for (int k0 = 0; k0 < 8; ++k0) {
      v2f av, bv;
      // A (16x4 f32): lanes 0-15 -> K={0,1}, lanes 16-31 -> K={2,3}; M = lane&15
      av.x = gamma_lds[k0 * 4 + kb + 0][m];
      av.y = gamma_lds[k0 * 4 + kb + 1][m];
      // B (4x16 f32): VGPR0 rows {kb}, VGPR1 rows {kb+1}; N = lane&15
      int gt = t0 + k0 * 4 + kb;
      const float* mp = mels + (size_t)(b * T2 + gt) * F + ftile * 16 + n;
      bv.x = mp[0];
      bv.y = mp[F];
      cacc = __builtin_amdgcn_wmma_f32_16x16x4_f32(
          /*neg_a=*/false, av, /*neg_b=*/false, bv,
          /*c_mod=*/(short)0, cacc, /*reuse_a=*/false, /*reuse_b=*/false);
    }
    __syncthreads();
  }

  // --- durations: reduce the 16 threads per s via LDS float atomics ---
  atomicAdd(&dur_lds[s_local], dur_acc);
  __syncthreads();

  if (tid < 16) {
    float tms = (text_mask[b * T1 + s0 + tid] != 0) ? 1.f : 0.f;
    out_dur[b * T1 + s0 + tid] = dur_lds[tid] * tms;
  }

  // --- write aligned mels: C layout VGPR r -> M=r (lanes 0-15) / r+8 (16-31) ---
#pragma unroll
  for (int r = 0; r < 8; ++r) {
    int M = r + (lane >> 4) * 8;
    int s = s0 + M;
    float tms = (text_mask[b * T1 + s] != 0) ? 1.f : 0.f;
    float val = cacc[r] / (dur_lds[M] + EPS) * tms;
    out_aligned[(size_t)(b * T1 + s) * F + ftile * 16 + n] = val;
  }
}

// ---------------------------------------------------------------------------
extern "C" void kernel_launch(void* const* d_in, const int* in_sizes, int n_in,
                              void* d_out, int out_size, void* d_ws, size_t ws_size,
                              hipStream_t stream) {
  const float* mels      = (const float*)d_in[0];
  const float* alpha     = (const float*)d_in[1];
  const int*   mel_mask  = (const int*)d_in[2];
  const int*   text_mask = (const int*)d_in[3];

  float* out = (float*)d_out;
  float* out_aligned = out;                         // [B, T1, F]
  float* out_dur     = out + (size_t)B * T1 * F;    // [B, T1]
  float* out_imv     = out_dur + (size_t)B * T1;    // [B, T2]

  float* ws   = (float*)d_ws;
  float* w_act = ws;                  // [B*T2]
  float* w_ivp = w_act + B * T2;      // [B*T2]
  float* w_imv = w_ivp + B * T2;      // [B*T2]
  float* w_m   = w_imv + B * T2;      // [B*T2]
  float* w_z   = w_m   + B * T2;      // [B*T2]

  const int rows = B * T2;            // 32768 rows, 8 waves/block
  dim3 blk256(256);

  act_prop_kernel<<<rows / 8, blk256, 0, stream>>>(mels, alpha, mel_mask, w_act, w_ivp);
  scan_kernel<<<B, dim3(32), 0, stream>>>(w_act, w_ivp, mel_mask, text_mask, w_imv, out_imv);
  stats_kernel<<<rows / 8, blk256, 0, stream>>>(w_imv, text_mask, w_m, w_z);
  align_gemm_kernel<<<B * 32, blk256, 0, stream>>>(mels, mel_mask, text_mask,
                                                   w_imv, w_m, w_z,
                                                   out_aligned, out_dur);
}